// GriddingReverse_20486994002219
// MI455X (gfx1250) — compile-verified
//
#include <hip/hip_runtime.h>

// GriddingReverse (GRNet rev_forward) on a 64^3 grid, batch 32.
// Memory-bound: ~134 MB HBM traffic -> ~5.7 us floor @ 23.3 TB/s.
// CDNA5 paths used:
//  - async global->LDS staging of z-slices (global_load_async_to_lds_b128,
//    ASYNCcnt, s_wait_asynccnt): grid read from HBM once, 8-way corner reuse
//    served from LDS (ds_load_b64 pairs, bank-conflict-free).
//  - non-temporal (TH=NT) output stores: the 100 MB write-once stream doesn't
//    evict the grid slices from L2, preserving cross-workgroup slice reuse.

#define DX 64
#define DY 64
#define DZ 64
#define PLANE (DX * DY)        // 4096 floats = 16 KB per z-slice
#define VOXELS (DX * DY * DZ)  // 262144

typedef float v4f __attribute__((ext_vector_type(4)));

__global__ __launch_bounds__(256)
void gridding_rev_kernel(const float* __restrict__ grid, float* __restrict__ out) {
    const int z   = blockIdx.x;   // 0..63
    const int b   = blockIdx.y;   // batch
    const int tid = threadIdx.x;  // 0..255

    float* out_plane = out + ((size_t)b * VOXELS + (size_t)z * PLANE) * 3;

    if (z == 0) {
        // Border plane: all-zero points (d_out is poisoned, must write zeros).
        // Plane base is 16B-aligned: vectorized NT fill, 3072 x b128 stores.
        v4f zero = {0.0f, 0.0f, 0.0f, 0.0f};
        v4f* p = (v4f*)out_plane;
        for (int i = tid; i < (PLANE * 3) / 4; i += 256)
            __builtin_nontemporal_store(zero, p + i);
        return;
    }

    __shared__ float s0[PLANE];  // slice z-1
    __shared__ float s1[PLANE];  // slice z

    const float* base0 = grid + ((size_t)b * VOXELS + (size_t)(z - 1) * PLANE);
    const float* base1 = base0 + PLANE;

    const unsigned lds0 = (unsigned)(size_t)(void*)s0;  // low 32 bits of flat addr == LDS offset
    const unsigned lds1 = (unsigned)(size_t)(void*)s1;

    // 16 KB per slice = 1024 x b128; 256 lanes x 4 chunks each, per slice.
    // GVS mode: lds_addr (VGPR), 32-bit byte offset (VGPR), uniform base (SGPR pair).
#pragma unroll
    for (int t = 0; t < 4; ++t) {
        const unsigned off = (unsigned)(tid + t * 256) * 16u;
        asm volatile("global_load_async_to_lds_b128 %0, %1, %2"
                     :: "v"(lds0 + off), "v"(off), "s"(base0) : "memory");
        asm volatile("global_load_async_to_lds_b128 %0, %1, %2"
                     :: "v"(lds1 + off), "v"(off), "s"(base1) : "memory");
    }
    asm volatile("s_wait_asynccnt 0" ::: "memory");
    __syncthreads();  // make all waves' async LDS writes visible to the workgroup

    const float rescale = 2.0f / 64.0f;  // final scaling 2/D
    const float halfD   = 32.0f;         // D/2

    for (int i = tid; i < PLANE; i += 256) {
        const int x = i & (DX - 1);
        const int y = i >> 6;
        float ox = 0.0f, oy = 0.0f, oz = 0.0f;
        if (x > 0 && y > 0) {
            const int i00 = (y - 1) * DX + (x - 1);
            const float c000 = s0[i00],      c001 = s0[i00 + 1];
            const float c010 = s0[i00 + DX], c011 = s0[i00 + DX + 1];
            const float c100 = s1[i00],      c101 = s1[i00 + 1];
            const float c110 = s1[i00 + DX], c111 = s1[i00 + DX + 1];

            const float wsum = c000 + c001 + c010 + c011 + c100 + c101 + c110 + c111;
            if (wsum > 0.0f) {
                const float inv = 1.0f / wsum;
                const float wx1 = c001 + c011 + c101 + c111;
                const float wy1 = c010 + c011 + c110 + c111;
                const float wz1 = c100 + c101 + c110 + c111;
                ox = (((float)(x - 1) * wsum + wx1) * inv - halfD) * rescale;
                oy = (((float)(y - 1) * wsum + wy1) * inv - halfD) * rescale;
                oz = (((float)(z - 1) * wsum + wz1) * inv - halfD) * rescale;
            }
        }
        // Stride-3 layout: three scalar NT stores (a 16B vector would race the
        // neighbor cell's first component; backend may fuse to b96 where aligned).
        __builtin_nontemporal_store(ox, &out_plane[i * 3 + 0]);
        __builtin_nontemporal_store(oy, &out_plane[i * 3 + 1]);
        __builtin_nontemporal_store(oz, &out_plane[i * 3 + 2]);
    }
}

extern "C" void kernel_launch(void* const* d_in, const int* in_sizes, int n_in,
                              void* d_out, int out_size, void* d_ws, size_t ws_size,
                              hipStream_t stream) {
    (void)n_in; (void)out_size; (void)d_ws; (void)ws_size;
    const float* grid = (const float*)d_in[0];
    float* out = (float*)d_out;
    const int batch = in_sizes[0] / VOXELS;  // 32 for the reference setup
    dim3 g(DZ, batch);
    gridding_rev_kernel<<<g, 256, 0, stream>>>(grid, out);
}